// SSIM_13125420057216
// MI455X (gfx1250) — compile-verified
//
#include <hip/hip_runtime.h>
#include <stdint.h>

// ---------------- geometry ----------------
#define IMG_H 384
#define IMG_W 512
#define OUT_H 382
#define OUT_W 510

#define TILE_W 64
#define TILE_H 16
#define IN_W   (TILE_W + 2)   // 66
#define IN_H   (TILE_H + 2)   // 18
#define LDS_ST 68             // padded row stride (floats)

#define BLOCK_THREADS 256

// ---------------- CDNA5 async global->LDS path (probe-confirmed signature) ----------------
#if defined(__has_builtin)
#  if __has_builtin(__builtin_amdgcn_global_load_async_to_lds_b32)
#    define HAVE_ASYNC_LDS 1
#  endif
#  if __has_builtin(__builtin_amdgcn_s_wait_asynccnt)
#    define HAVE_WAIT_ASYNC 1
#  endif
#  if __has_builtin(__builtin_nontemporal_store)
#    define HAVE_NT_STORE 1
#  endif
#endif

typedef __attribute__((address_space(1))) int gbl_i32;   // global int*
typedef __attribute__((address_space(3))) int lds_i32;   // LDS int*

__device__ __forceinline__ void copy_g2l_b32(const float* g, float* l) {
#ifdef HAVE_ASYNC_LDS
  // (global src, lds dst, imm offset, imm cpol)
  __builtin_amdgcn_global_load_async_to_lds_b32(
      (gbl_i32*)(void*)const_cast<float*>(g),
      (lds_i32*)(void*)l,
      0, 0);
#else
  *l = *g;
#endif
}

__device__ __forceinline__ void wait_async_lds() {
#ifdef HAVE_ASYNC_LDS
#  ifdef HAVE_WAIT_ASYNC
  __builtin_amdgcn_s_wait_asynccnt(0);
#  else
  asm volatile("s_wait_asynccnt 0" ::: "memory");
#  endif
#endif
}

__device__ __forceinline__ void store_nt(float* p, float v) {
#ifdef HAVE_NT_STORE
  __builtin_nontemporal_store(v, p);
#else
  *p = v;
#endif
}

// ---------------- kernel ----------------
__global__ __launch_bounds__(BLOCK_THREADS)
void ssim_map_kernel(const float* __restrict__ x,
                     const float* __restrict__ y,
                     float* __restrict__ out) {
  __shared__ float xt[IN_H * LDS_ST];
  __shared__ float yt[IN_H * LDS_ST];

  const int tile_x = blockIdx.x;           // 0..7
  const int tile_y = blockIdx.y;           // 0..23
  const int plane  = blockIdx.z;           // 0..95
  const int row0   = tile_y * TILE_H;      // input & output row origin
  const int col0   = tile_x * TILE_W;      // input & output col origin

  const float* __restrict__ xp = x + (size_t)plane * (IMG_H * IMG_W);
  const float* __restrict__ yp = y + (size_t)plane * (IMG_H * IMG_W);

  const int tid = threadIdx.x;
  const int wv  = tid >> 5;                // wave id 0..7
  const int ln  = tid & 31;                // lane  0..31

  // ---- Stage 1: async-stage the 66x18 halo tiles of x and y into LDS ----
  // Wave w owns input-tile rows {w, w+8, w+16 (w<2)}; lanes cover columns
  // {ln, 32+ln, 64+ln (ln<2)}. No division, fully unrolled, straight-line
  // global_load_async_to_lds_b32 issues tracked by ASYNCcnt.
#pragma unroll
  for (int rr = 0; rr < 3; ++rr) {
    const int r = wv + rr * 8;             // tile row
    if (rr < 2 || wv < 2) {                // r < 18
      int gr = row0 + r; gr = gr > (IMG_H - 1) ? (IMG_H - 1) : gr;  // edge clamp
      const float* xrow = xp + gr * IMG_W;
      const float* yrow = yp + gr * IMG_W;
      float* xl = &xt[r * LDS_ST];
      float* yl = &yt[r * LDS_ST];

      const int c0 = ln;                   // cols 0..31   (gc <= 479, in range)
      copy_g2l_b32(xrow + col0 + c0, xl + c0);
      copy_g2l_b32(yrow + col0 + c0, yl + c0);

      const int c1 = 32 + ln;              // cols 32..63  (gc <= 511, in range)
      copy_g2l_b32(xrow + col0 + c1, xl + c1);
      copy_g2l_b32(yrow + col0 + c1, yl + c1);

      if (ln < 2) {                        // halo cols 64..65
        const int c2 = 64 + ln;
        int gc = col0 + c2; gc = gc > (IMG_W - 1) ? (IMG_W - 1) : gc;  // edge clamp
        copy_g2l_b32(xrow + gc, xl + c2);
        copy_g2l_b32(yrow + gc, yl + c2);
      }
    }
  }
  wait_async_lds();      // per-wave: all async writes landed in LDS
  __syncthreads();       // workgroup: everyone's tile data visible

  // ---- Stage 2: separable 3x3 box sums + SSIM math ----
  // Thread -> (column lc, 4 consecutive output rows starting at lr0).
  const int lc  = tid & (TILE_W - 1);        // 0..63
  const int lr0 = (tid >> 6) * 4;            // 0,4,8,12

  float rsx[6], rsy[6], rsxx[6], rsyy[6], rsxy[6];
#pragma unroll
  for (int k = 0; k < 6; ++k) {
    const int r = lr0 + k;                   // input tile row, <= 17
    const float x0 = xt[r * LDS_ST + lc + 0];
    const float x1 = xt[r * LDS_ST + lc + 1];
    const float x2 = xt[r * LDS_ST + lc + 2];
    const float y0 = yt[r * LDS_ST + lc + 0];
    const float y1 = yt[r * LDS_ST + lc + 1];
    const float y2 = yt[r * LDS_ST + lc + 2];
    rsx[k]  = x0 + x1 + x2;
    rsy[k]  = y0 + y1 + y2;
    rsxx[k] = x0 * x0 + x1 * x1 + x2 * x2;
    rsyy[k] = y0 * y0 + y1 * y1 + y2 * y2;
    rsxy[k] = x0 * y0 + x1 * y1 + x2 * y2;
  }

  const float C1   = 1.0e-4f;  // 0.01^2
  const float C2   = 9.0e-4f;  // 0.03^2
  const float inv9 = 1.0f / 9.0f;

  float* __restrict__ op = out + (size_t)plane * (OUT_H * OUT_W);

#pragma unroll
  for (int j = 0; j < 4; ++j) {
    const int out_r = row0 + lr0 + j;
    const int out_c = col0 + lc;
    if (out_r < OUT_H && out_c < OUT_W) {
      const float Sx  = rsx[j]  + rsx[j + 1]  + rsx[j + 2];
      const float Sy  = rsy[j]  + rsy[j + 1]  + rsy[j + 2];
      const float Sxx = rsxx[j] + rsxx[j + 1] + rsxx[j + 2];
      const float Syy = rsyy[j] + rsyy[j + 1] + rsyy[j + 2];
      const float Sxy = rsxy[j] + rsxy[j + 1] + rsxy[j + 2];

      const float mux   = Sx * inv9;
      const float muy   = Sy * inv9;
      const float mu_xy = mux * muy;
      const float mu_xx = mux * mux;
      const float mu_yy = muy * muy;
      const float sig_x  = Sxx * inv9 - mu_xx;
      const float sig_y  = Syy * inv9 - mu_yy;
      const float sig_xy = Sxy * inv9 - mu_xy;

      const float numer = (2.0f * mu_xy + C1) * (2.0f * sig_xy + C2);
      const float denom = (mu_xx + mu_yy + C1) * (sig_x + sig_y + C2);
      float v = (1.0f - numer / denom) * 0.5f;
      v = v < 0.0f ? 0.0f : v;
      v = v > 1.0f ? 1.0f : v;
      store_nt(&op[out_r * OUT_W + out_c], v);
    }
  }
}

// ---------------- launch ----------------
extern "C" void kernel_launch(void* const* d_in, const int* in_sizes, int n_in,
                              void* d_out, int out_size, void* d_ws, size_t ws_size,
                              hipStream_t stream) {
  const float* x = (const float*)d_in[0];
  const float* y = (const float*)d_in[1];
  float* out = (float*)d_out;

  const int planes = in_sizes[0] / (IMG_H * IMG_W);  // 32*3 = 96

  dim3 grid((OUT_W + TILE_W - 1) / TILE_W,   // 8
            (OUT_H + TILE_H - 1) / TILE_H,   // 24
            planes);                          // 96
  ssim_map_kernel<<<grid, BLOCK_THREADS, 0, stream>>>(x, y, out);
}